// Net_skip_11390253269711
// MI455X (gfx1250) — compile-verified
//
#include <hip/hip_runtime.h>
#include <hip/hip_bf16.h>

typedef float v2f __attribute__((ext_vector_type(2)));
typedef float v8f __attribute__((ext_vector_type(8)));

__device__ __forceinline__ v8f wmma4(v2f a, v2f b, v8f c) {
    // D = A(16x4,f32) x B(4x16,f32) + C(16x16,f32)
    return __builtin_amdgcn_wmma_f32_16x16x4_f32(
        /*neg_a=*/false, a, /*neg_b=*/false, b,
        /*c_mod=*/(short)0, c, /*reuse_a=*/false, /*reuse_b=*/false);
}

// ---------------- degree / normalization ----------------
__global__ void gcn_init_deg(float* deg, int n) {
    int i = blockIdx.x * blockDim.x + threadIdx.x;
    if (i < n) deg[i] = 1.0f;   // self loop
}

__global__ void gcn_deg_edges(const int* __restrict__ dst, float* deg, int e) {
    int i = blockIdx.x * blockDim.x + threadIdx.x;
    if (i < e) atomicAdd(&deg[dst[i]], 1.0f);
}

__global__ void gcn_isd(const float* __restrict__ deg, float* isd, int n) {
    int i = blockIdx.x * blockDim.x + threadIdx.x;
    if (i < n) isd[i] = rsqrtf(deg[i]);
}

// ---------------- node MLP + first transform (WMMA) ----------------
// h = relu(x@fc1+b1); h = relu(h@fc2+b2); t1 = concat(h,x)[N,19] @ w1[19,29]
__global__ __launch_bounds__(256) void gcn_mlp_t1(
    const float* __restrict__ x,
    const float* __restrict__ fc1_w, const float* __restrict__ fc1_b,
    const float* __restrict__ fc2_w, const float* __restrict__ fc2_b,
    const float* __restrict__ w1,
    float* __restrict__ t1, int n_nodes)
{
    __shared__ float tile[8][16][17];   // per-wave 16x16 staging (padded)
    const int lane = threadIdx.x & 31;
    const int wave = threadIdx.x >> 5;
    const int half = lane >> 4;       // 0 or 1
    const int lm   = lane & 15;       // m (A/C row) or n (B/C col)
    const int base = (blockIdx.x * 8 + wave) * 16;
    int myNode = base + lm; if (myNode >= n_nodes) myNode = n_nodes - 1;

    // ---- fc1: A = x[16x3 -> 16x4], B = fc1_w[3x16 -> 4x16] ----
    v8f c = {};
    {
        const int k0 = 2 * half;
        v2f a, b;
        a[0] = (k0 + 0 < 3) ? x[myNode * 3 + k0 + 0] : 0.f;
        a[1] = (k0 + 1 < 3) ? x[myNode * 3 + k0 + 1] : 0.f;
        b[0] = (k0 + 0 < 3) ? fc1_w[(k0 + 0) * 16 + lm] : 0.f;
        b[1] = (k0 + 1 < 3) ? fc1_w[(k0 + 1) * 16 + lm] : 0.f;
        c = wmma4(a, b, c);
    }
    {   // bias + relu, stage to LDS (C layout -> row/col tile)
        float bias = fc1_b[lm];
        #pragma unroll
        for (int i = 0; i < 8; ++i)
            tile[wave][i + 8 * half][lm] = fmaxf(c[i] + bias, 0.f);
    }
    __syncthreads();

    // ---- fc2: 16x16 @ 16x16, K = 4 chunks of 4 ----
    v8f c2 = {};
    #pragma unroll
    for (int j = 0; j < 4; ++j) {
        const int k0 = 4 * j + 2 * half;
        v2f a, b;
        a[0] = tile[wave][lm][k0 + 0];
        a[1] = tile[wave][lm][k0 + 1];
        b[0] = fc2_w[(k0 + 0) * 16 + lm];
        b[1] = fc2_w[(k0 + 1) * 16 + lm];
        c2 = wmma4(a, b, c2);
    }
    __syncthreads();
    {
        float bias = fc2_b[lm];
        #pragma unroll
        for (int i = 0; i < 8; ++i)
            tile[wave][i + 8 * half][lm] = fmaxf(c2[i] + bias, 0.f);
    }
    __syncthreads();

    // ---- t1 = concat(h16, x3)[16x19 -> 16x20] @ w1[19,29 -> 20x32] ----
    v8f d0 = {}, d1 = {};
    #pragma unroll
    for (int j = 0; j < 5; ++j) {
        v2f a, b0, b1;
        #pragma unroll
        for (int jj = 0; jj < 2; ++jj) {
            const int k = 4 * j + 2 * half + jj;
            float av;
            if (k < 16)      av = tile[wave][lm][k];
            else if (k < 19) av = x[myNode * 3 + (k - 16)];
            else             av = 0.f;
            a[jj]  = av;
            b0[jj] = (k < 19) ? w1[k * 29 + lm] : 0.f;
            b1[jj] = (k < 19 && lm + 16 < 29) ? w1[k * 29 + lm + 16] : 0.f;
        }
        d0 = wmma4(a, b0, d0);
        d1 = wmma4(a, b1, d1);
    }
    #pragma unroll
    for (int i = 0; i < 8; ++i) {
        const int node = base + i + 8 * half;
        if (node < n_nodes) {
            t1[node * 32 + lm] = d0[i];
            if (lm + 16 < 29) t1[node * 32 + lm + 16] = d1[i];
        }
    }
}

// ---------------- post-aggregation transform (WMMA) ----------------
// h = relu(agg[:, :29] + b); t = concat(h, x)[N,32] @ W[32,29]
__global__ __launch_bounds__(256) void gcn_post_t(
    const float* __restrict__ agg, const float* __restrict__ bvec,
    const float* __restrict__ x,   const float* __restrict__ W,
    float* __restrict__ t, int n_nodes)
{
    const int lane = threadIdx.x & 31;
    const int wave = threadIdx.x >> 5;
    const int half = lane >> 4;
    const int lm   = lane & 15;
    const int base = (blockIdx.x * 8 + wave) * 16;
    int myNode = base + lm; if (myNode >= n_nodes) myNode = n_nodes - 1;

    v8f d0 = {}, d1 = {};
    #pragma unroll
    for (int j = 0; j < 8; ++j) {          // K = 32
        v2f a, b0, b1;
        #pragma unroll
        for (int jj = 0; jj < 2; ++jj) {
            const int k = 4 * j + 2 * half + jj;
            float av;
            if (k < 29) av = fmaxf(agg[myNode * 32 + k] + bvec[k], 0.f);
            else        av = x[myNode * 3 + (k - 29)];
            a[jj]  = av;
            b0[jj] = W[k * 29 + lm];
            b1[jj] = (lm + 16 < 29) ? W[k * 29 + lm + 16] : 0.f;
        }
        d0 = wmma4(a, b0, d0);
        d1 = wmma4(a, b1, d1);
    }
    #pragma unroll
    for (int i = 0; i < 8; ++i) {
        const int node = base + i + 8 * half;
        if (node < n_nodes) {
            t[node * 32 + lm] = d0[i];
            if (lm + 16 < 29) t[node * 32 + lm + 16] = d1[i];
        }
    }
}

// ---------------- aggregation ----------------
// seed agg with self-loop: agg[i,f] = t[i,f] / deg[i]  (f < 29), 0 otherwise
__global__ void gcn_seed_agg(const float* __restrict__ t, const float* __restrict__ deg,
                             float* __restrict__ agg, int n) {
    int i = blockIdx.x * blockDim.x + threadIdx.x;
    int node = i >> 5, f = i & 31;
    if (node >= n) return;
    agg[node * 32 + f] = (f < 29) ? t[node * 32 + f] / deg[node] : 0.f;
}

// one wave per edge; lanes 0..28 handle features (coalesced gather + atomics)
__global__ void gcn_edge_agg29(const int* __restrict__ src, const int* __restrict__ dst,
                               const float* __restrict__ isd, const float* __restrict__ t,
                               float* __restrict__ agg, int n_edges)
{
    int gid = blockIdx.x * blockDim.x + threadIdx.x;
    int e = gid >> 5, f = gid & 31;
    if (e >= n_edges) return;
    int s = src[e], d = dst[e];
    float norm = isd[s] * isd[d];
    if (f < 29) atomicAdd(&agg[d * 32 + f], t[s * 32 + f] * norm);
}

// ---------------- final layer ----------------
// t3[i] = concat(relu(agg2+b2), x)[32] . w3[32]
__global__ void gcn_t3(const float* __restrict__ agg2, const float* __restrict__ b2,
                       const float* __restrict__ x, const float* __restrict__ w3,
                       float* __restrict__ t3, int n)
{
    int i = blockIdx.x * blockDim.x + threadIdx.x;
    if (i >= n) return;
    float acc = 0.f;
    #pragma unroll
    for (int k = 0; k < 29; ++k) acc += fmaxf(agg2[i * 32 + k] + b2[k], 0.f) * w3[k];
    #pragma unroll
    for (int k = 29; k < 32; ++k) acc += x[i * 3 + (k - 29)] * w3[k];
    t3[i] = acc;
}

__global__ void gcn_init_out(const float* __restrict__ t3, const float* __restrict__ deg,
                             const float* __restrict__ b3, float* __restrict__ out, int n) {
    int i = blockIdx.x * blockDim.x + threadIdx.x;
    if (i < n) out[i] = b3[0] + t3[i] / deg[i];
}

__global__ void gcn_edge_agg1(const int* __restrict__ src, const int* __restrict__ dst,
                              const float* __restrict__ isd, const float* __restrict__ t3,
                              float* __restrict__ out, int n_edges)
{
    int e = blockIdx.x * blockDim.x + threadIdx.x;
    if (e >= n_edges) return;
    int s = src[e], d = dst[e];
    atomicAdd(&out[d], t3[s] * isd[s] * isd[d]);
}

extern "C" void kernel_launch(void* const* d_in, const int* in_sizes, int n_in,
                              void* d_out, int out_size, void* d_ws, size_t ws_size,
                              hipStream_t stream) {
    const float* x     = (const float*)d_in[0];
    const int*   ei    = (const int*)d_in[1];
    const float* fc1_w = (const float*)d_in[2];
    const float* fc1_b = (const float*)d_in[3];
    const float* fc2_w = (const float*)d_in[4];
    const float* fc2_b = (const float*)d_in[5];
    const float* w1    = (const float*)d_in[6];
    const float* b1    = (const float*)d_in[7];
    const float* w2    = (const float*)d_in[8];
    const float* b2    = (const float*)d_in[9];
    const float* w3    = (const float*)d_in[10];
    const float* b3    = (const float*)d_in[11];

    const int N = in_sizes[0] / 3;
    const int E = in_sizes[1] / 2;
    const int* src = ei;
    const int* dst = ei + E;

    float* ws   = (float*)d_ws;
    float* deg  = ws;                  // N
    float* isd  = ws + (size_t)N;      // N
    float* t3   = ws + (size_t)2 * N;  // N
    float* bufT = ws + (size_t)3 * N;  // 32*N (t1 / t2, stride 32)
    float* bufA = ws + (size_t)35 * N; // 32*N (agg1 / agg2, stride 32)

    float* out = (float*)d_out;

    const int TB = 256;
    const int gN   = (N + TB - 1) / TB;
    const int gE   = (E + TB - 1) / TB;
    const int gN32 = (int)(((long long)N * 32 + TB - 1) / TB);
    const int gE32 = (int)(((long long)E * 32 + TB - 1) / TB);
    const int gTile = (N + 127) / 128;   // 8 waves/block, 16 nodes/wave

    // degrees + normalization
    gcn_init_deg<<<gN, TB, 0, stream>>>(deg, N);
    gcn_deg_edges<<<gE, TB, 0, stream>>>(dst, deg, E);
    gcn_isd<<<gN, TB, 0, stream>>>(deg, isd, N);

    // node MLP + transform 1 (WMMA)
    gcn_mlp_t1<<<gTile, TB, 0, stream>>>(x, fc1_w, fc1_b, fc2_w, fc2_b, w1, bufT, N);

    // GCN layer 1 aggregation
    gcn_seed_agg<<<gN32, TB, 0, stream>>>(bufT, deg, bufA, N);
    gcn_edge_agg29<<<gE32, TB, 0, stream>>>(src, dst, isd, bufT, bufA, E);

    // relu(agg+b1), concat x, @ w2 (WMMA)
    gcn_post_t<<<gTile, TB, 0, stream>>>(bufA, b1, x, w2, bufT, N);

    // GCN layer 2 aggregation
    gcn_seed_agg<<<gN32, TB, 0, stream>>>(bufT, deg, bufA, N);
    gcn_edge_agg29<<<gE32, TB, 0, stream>>>(src, dst, isd, bufT, bufA, E);

    // final: relu(agg+b2), concat x, @ w3 -> t3; out = b3 + aggregate(t3)
    gcn_t3<<<gN, TB, 0, stream>>>(bufA, b2, x, w3, t3, N);
    gcn_init_out<<<gN, TB, 0, stream>>>(t3, deg, b3, out, N);
    gcn_edge_agg1<<<gE, TB, 0, stream>>>(src, dst, isd, t3, out, E);
}